// backFEM_2D_35871566856970
// MI455X (gfx1250) — compile-verified
//
#include <hip/hip_runtime.h>

typedef float v2f __attribute__((ext_vector_type(2)));
typedef float v8f __attribute__((ext_vector_type(8)));
typedef float f4  __attribute__((ext_vector_type(4)));

// ---------------------------------------------------------------------------
// Kernel 1: zero the dense output (339.7 MB). Pure streaming store problem:
// wide B128 non-temporal stores (output >> 192MB L2, so NT is the right hint).
// ---------------------------------------------------------------------------
__global__ void fem_zero_kernel(float* __restrict__ out, long long total) {
    long long i = (long long)blockIdx.x * blockDim.x + threadIdx.x;
    long long count4 = total >> 2;
    if (i < count4) {
        f4 z = {0.0f, 0.0f, 0.0f, 0.0f};
        __builtin_nontemporal_store(z, (f4*)out + i);
    }
    // generic tail (total % 4 elements), zero for this problem size
    long long tail = total & 3;
    if (i < tail) out[count4 * 4 + i] = 0.0f;
}

// ---------------------------------------------------------------------------
// Kernel 2: element-matrix assembly via V_WMMA_F32_16X16X4_F32.
//
// For triangle t with vertices p0,p1,p2:
//   det  = (x1-x0)(y2-y0) - (x2-x0)(y1-y0),  area = |det|/2
//   g0 = ((y1-y2), (x2-x1))/det, g1 = ((y2-y0), (x0-x2))/det,
//   g2 = ((y0-y1), (x1-x0))/det
//   M_elem[a][b] = area * (g_a . g_b)  ==  (area*G) * G^T   (G is 3x2)
//
// One wave packs 5 triangles: A(16x4) rows 3q..3q+2 = area*G of triangle q
// (K=2,3 zero, row 15 zero); B(4x16) cols likewise = G^T. One WMMA produces
// all 5 element matrices as the diagonal 3x3 blocks of D(16x16).
//
// A layout (32-bit, 16x4): lanes 0-15 hold row M=lane, VGPR0=K0, VGPR1=K1;
// lanes 16-31 hold K2,K3 (zero here). B layout mirrors with N=lane%16.
// D layout: VGPR v, lane l -> (M = 8*(l>>4) + v, N = l&15).
// ---------------------------------------------------------------------------
__global__ void fem_assemble_kernel(const float* __restrict__ pts,
                                    const int*   __restrict__ cells,
                                    float*       __restrict__ out,
                                    int T, int N) {
    const int lane = threadIdx.x & 31;
    const int wave = blockIdx.x * (blockDim.x >> 5) + (threadIdx.x >> 5);
    const int nwaves = (T + 4) / 5;
    if (wave >= nwaves) return;          // wave-uniform: EXEC stays all-1s below

    const long long tri_base = (long long)wave * 5;

    // Per-lane operand setup: lane m (0..14) holds basis r = m%3 of triangle
    // tri_base + m/3. Lanes 15..31 carry zeros (unused row / K=2,3).
    float ax = 0.0f, ay = 0.0f, bx = 0.0f, by = 0.0f;
    if (lane < 15) {
        long long t = tri_base + lane / 3;
        if (t < T) {
            const int r  = lane % 3;
            const int i0 = cells[3 * t + 0];
            const int i1 = cells[3 * t + 1];
            const int i2 = cells[3 * t + 2];
            const float x0 = pts[2 * i0], y0 = pts[2 * i0 + 1];
            const float x1 = pts[2 * i1], y1 = pts[2 * i1 + 1];
            const float x2 = pts[2 * i2], y2 = pts[2 * i2 + 1];
            const float det = (x1 - x0) * (y2 - y0) - (x2 - x0) * (y1 - y0);
            const float inv = 1.0f / det;
            float gx, gy;
            if (r == 0)      { gx = (y1 - y2) * inv; gy = (x2 - x1) * inv; }
            else if (r == 1) { gx = (y2 - y0) * inv; gy = (x0 - x2) * inv; }
            else             { gx = (y0 - y1) * inv; gy = (x1 - x0) * inv; }
            const float area = 0.5f * fabsf(det);
            ax = area * gx;  ay = area * gy;   // A row: area-scaled gradient
            bx = gx;         by = gy;          // B col: gradient
        }
    }
    // EXEC reconverged: all 32 lanes active for the WMMA (ISA requirement).
    v2f a; a.x = ax; a.y = ay;
    v2f b; b.x = bx; b.y = by;
    v8f c = {};
    v8f d = __builtin_amdgcn_wmma_f32_16x16x4_f32(
        /*neg_a=*/false, a, /*neg_b=*/false, b,
        /*c_mod=*/(short)0, c, /*reuse_a=*/false, /*reuse_b=*/false);

    // Extract the 5 diagonal 3x3 blocks and scatter-add -M (output is -M).
    const int n    = lane & 15;
    const int half = lane >> 4;
    if (n < 15) {
        const int q = n / 3;
        const long long t = tri_base + q;
        if (t < T) {
            const int col = cells[3 * t + (n - 3 * q)];
#pragma unroll
            for (int v = 0; v < 8; ++v) {
                const int m = half * 8 + v;           // D row for this VGPR
                if (m >= 3 * q && m < 3 * q + 3) {    // inside diagonal block
                    const int row = cells[3 * t + (m - 3 * q)];
                    atomicAdd(&out[(size_t)row * N + col], -d[v]);
                }
            }
        }
    }
}

extern "C" void kernel_launch(void* const* d_in, const int* in_sizes, int n_in,
                              void* d_out, int out_size, void* d_ws, size_t ws_size,
                              hipStream_t stream) {
    const float* pts   = (const float*)d_in[0];   // mesh_points  (N x 2)
    const int*   cells = (const int*)d_in[1];     // cell_node_map (T x 3)
    float*       out   = (float*)d_out;           // dense N x N, float32

    const int N = in_sizes[0] / 2;                // number of mesh nodes
    const int T = in_sizes[1] / 3;                // number of triangles

    // Pass 1: zero-fill (bandwidth-bound; ~14.6 us floor at 23.3 TB/s).
    const long long total  = (long long)out_size;
    const long long count4 = (total + 3) >> 2;
    const int threads = 256;
    const long long blocks0 = (count4 + threads - 1) / threads;
    fem_zero_kernel<<<(int)blocks0, threads, 0, stream>>>(out, total);

    // Pass 2: WMMA element assembly + atomic scatter (5 triangles / wave).
    const int nwaves        = (T + 4) / 5;
    const int wavesPerBlock = 8;                  // 256 threads, wave32
    const int blocks1 = (nwaves + wavesPerBlock - 1) / wavesPerBlock;
    fem_assemble_kernel<<<blocks1, threads, 0, stream>>>(pts, cells, out, T, N);
}